// GCNEdgePredictor_55972013801705
// MI455X (gfx1250) — compile-verified
//
#include <hip/hip_runtime.h>
#include <hip/hip_bf16.h>

// Problem constants (match the reference file).
#define N_NODES 100000
#define N_EDGES 1000000
#define DIN  64
#define DHID 64
#define DOUTF 32

typedef __attribute__((ext_vector_type(2))) float v2f;
typedef __attribute__((ext_vector_type(8))) float v8f;

// ---------------------------------------------------------------------------
// deg[i] = 1 (self-loop weight), then accumulate edge weights at dst.
// ---------------------------------------------------------------------------
__global__ void k_init_deg(float* __restrict__ deg, int n) {
    int i = blockIdx.x * blockDim.x + threadIdx.x;
    if (i < n) deg[i] = 1.0f;
}

__global__ void k_accum_deg(const int* __restrict__ dst, const float* __restrict__ ew,
                            float* __restrict__ deg, int e) {
    int i = blockIdx.x * blockDim.x + threadIdx.x;
    if (i < e) atomicAdd(&deg[dst[i]], ew[i]);
}

// deg -> dinv in place; deg >= 1 always (self loops), so rsqrt is safe.
__global__ void k_rsqrt_inplace(float* __restrict__ deg, int n) {
    int i = blockIdx.x * blockDim.x + threadIdx.x;
    if (i < n) deg[i] = rsqrtf(deg[i]);
}

// out[i*F + j] = b[j]  (seed accumulators with the bias)
template <int F>
__global__ void k_init_bias(float* __restrict__ out, const float* __restrict__ b, int total) {
    int i = blockIdx.x * blockDim.x + threadIdx.x;
    if (i < total) out[i] = b[i & (F - 1)];
}

// ---------------------------------------------------------------------------
// C[M x NDIM] = (RELU ? max(A,0) : A)[M x KDIM]  @  B[KDIM x NDIM]
// One wave per 16x16 tile of C, using V_WMMA_F32_16X16X4_F32 over K.
//   blockDim.x = (NDIM/16)*32, wave w handles column tile w, block handles 16 rows.
// A fragment (16x4 f32): lanes 0-15 -> {K=k,k+1}, lanes 16-31 -> {K=k+2,k+3}.
// B fragment (4x16 f32): mirrored, N striped across lanes.
// C/D (16x16 f32, 8 VGPRs): VGPR r -> M = r (lanes 0-15) / 8+r (lanes 16-31),
//                           N = lane & 15.
// ---------------------------------------------------------------------------
template <int KDIM, int NDIM, bool RELU>
__global__ void k_gemm_wmma(const float* __restrict__ A, const float* __restrict__ B,
                            float* __restrict__ C) {
    const int lane  = threadIdx.x & 31;
    const int wave  = threadIdx.x >> 5;
    const int row0  = blockIdx.x * 16;
    const int col0  = wave * 16;
    const int mrow  = lane & 15;           // row-in-tile for A, col-in-tile for B
    const int khalf = (lane >> 4) * 2;     // 0 for lanes 0-15, 2 for lanes 16-31

    v8f acc = {};
#pragma unroll
    for (int k = 0; k < KDIM; k += 4) {
        const float* ap = A + (size_t)(row0 + mrow) * KDIM + (k + khalf);
        v2f a;
        a.x = ap[0];
        a.y = ap[1];
        if (RELU) { a.x = fmaxf(a.x, 0.0f); a.y = fmaxf(a.y, 0.0f); }

        const float* bp = B + (size_t)(k + khalf) * NDIM + (col0 + mrow);
        v2f bfr;
        bfr.x = bp[0];      // K = k+khalf
        bfr.y = bp[NDIM];   // K = k+khalf+1

        acc = __builtin_amdgcn_wmma_f32_16x16x4_f32(
            /*neg_a=*/false, a, /*neg_b=*/false, bfr,
            /*c_mod=*/(short)0, acc, /*reuse_a=*/false, /*reuse_b=*/false);
    }

    const int mbase = row0 + ((lane >> 4) << 3);
    const int col   = col0 + (lane & 15);
#pragma unroll
    for (int r = 0; r < 8; ++r)
        C[(size_t)(mbase + r) * NDIM + col] = acc[r];
}

// ---------------------------------------------------------------------------
// Edge aggregation: out[dst] += h[src] * (dinv[src]*w*dinv[dst])
// 32 lanes per edge; e in [nE, nTot) are the implicit self-loops (w = 1).
// h and out are L2-resident (25.6 MB / 12.8 MB << 192 MB L2), so the random
// gathers and f32 atomics run at L2 bandwidth; only the edge lists stream
// from HBM.
// ---------------------------------------------------------------------------
template <int F>
__global__ void k_scatter(const float* __restrict__ h,
                          const int* __restrict__ src, const int* __restrict__ dst,
                          const float* __restrict__ ew, const float* __restrict__ dinv,
                          float* __restrict__ out, int nE, int nTot) {
    int gid  = blockIdx.x * blockDim.x + threadIdx.x;
    int e    = gid >> 5;
    int lane = gid & 31;
    if (e >= nTot) return;

    int s, d; float w;
    if (e < nE) { s = src[e]; d = dst[e]; w = ew[e]; }
    else        { s = d = e - nE; w = 1.0f; }

    float nrm = dinv[s] * w * dinv[d];
    const float* hs = h + (size_t)s * F;
    float* od = out + (size_t)d * F;
#pragma unroll
    for (int j = lane; j < F; j += 32)
        atomicAdd(&od[j], hs[j] * nrm);
}

// ---------------------------------------------------------------------------
extern "C" void kernel_launch(void* const* d_in, const int* in_sizes, int n_in,
                              void* d_out, int out_size, void* d_ws, size_t ws_size,
                              hipStream_t stream) {
    (void)in_sizes; (void)n_in; (void)out_size; (void)ws_size;

    const float* x   = (const float*)d_in[0];            // [N, 64]
    const int*   ei  = (const int*)  d_in[1];            // [2, E]
    const float* ew  = (const float*)d_in[2];            // [E]
    const float* W1  = (const float*)d_in[3];            // [64, 64]
    const float* b1  = (const float*)d_in[4];            // [64]
    const float* W2  = (const float*)d_in[5];            // [64, 32]
    const float* b2  = (const float*)d_in[6];            // [32]
    const int* src = ei;
    const int* dst = ei + N_EDGES;
    float* out = (float*)d_out;                          // [N, 32]

    // Workspace: dinv (N) | h1 (N*64) | a1 (N*64); h2 aliases h1 (dead then).
    float* dinv = (float*)d_ws;
    float* h1   = dinv + N_NODES;
    float* a1   = h1 + (size_t)N_NODES * DHID;
    float* h2   = h1;                                    // reuse, [N, 32]

    const int nTot = N_EDGES + N_NODES;                  // edges + self-loops
    const int TPB  = 256;

    // --- degree / normalization -------------------------------------------
    k_init_deg<<<(N_NODES + TPB - 1) / TPB, TPB, 0, stream>>>(dinv, N_NODES);
    k_accum_deg<<<(N_EDGES + TPB - 1) / TPB, TPB, 0, stream>>>(dst, ew, dinv, N_EDGES);
    k_rsqrt_inplace<<<(N_NODES + TPB - 1) / TPB, TPB, 0, stream>>>(dinv, N_NODES);

    // --- layer 1: h1 = x @ W1 (WMMA f32), a1 = b1 + scatter ----------------
    k_gemm_wmma<DIN, DHID, false>
        <<<N_NODES / 16, (DHID / 16) * 32, 0, stream>>>(x, W1, h1);
    k_init_bias<DHID>
        <<<(N_NODES * DHID + TPB - 1) / TPB, TPB, 0, stream>>>(a1, b1, N_NODES * DHID);
    {
        long long thr = (long long)nTot * 32;
        k_scatter<DHID>
            <<<(unsigned)((thr + TPB - 1) / TPB), TPB, 0, stream>>>(
                h1, src, dst, ew, dinv, a1, N_EDGES, nTot);
    }

    // --- layer 2: h2 = relu(a1) @ W2 (relu fused into A load) --------------
    k_gemm_wmma<DHID, DOUTF, true>
        <<<N_NODES / 16, (DOUTF / 16) * 32, 0, stream>>>(a1, W2, h2);
    k_init_bias<DOUTF>
        <<<(N_NODES * DOUTF + TPB - 1) / TPB, TPB, 0, stream>>>(out, b2, N_NODES * DOUTF);
    {
        long long thr = (long long)nTot * 32;
        k_scatter<DOUTF>
            <<<(unsigned)((thr + TPB - 1) / TPB), TPB, 0, stream>>>(
                h2, src, dst, ew, dinv, out, N_EDGES, nTot);
    }
}